// GIN_SIA_16630113370112
// MI455X (gfx1250) — compile-verified
//
#include <hip/hip_runtime.h>
#include <hip/hip_bf16.h>

// GIN GNN for MI455X (gfx1250). fp32 WMMA (V_WMMA_F32_16X16X4_F32) for all
// node-feature GEMMs (templated on mode+K so the K-loop fully unrolls and the
// compiler can pipeline loads across WMMAs); weight panel staged in LDS;
// float atomics for edge scatter + pooling; BN stats via atomic column
// reductions, BN+ReLU apply fused into the consumer GEMM.

typedef __attribute__((ext_vector_type(2))) float v2f;
typedef __attribute__((ext_vector_type(8))) float v8f;

#define HID 64
#define INFO 32
#define XCOLS 96   // HID + INFO
#define NGRAPH 512
#define TARGET 10
#define BN_EPS 1e-5f

// ---------------------------------------------------------------------------
// Fill xin[:,64:96] with struc_info[batch[i]] and histogram node counts.
__global__ void k_init_rep_counts(const int* __restrict__ batch,
                                  const float* __restrict__ struc,
                                  float* __restrict__ xin,
                                  int* __restrict__ counts, int N) {
  int i = blockIdx.x * blockDim.x + threadIdx.x;
  if (i >= N) return;
  int g = batch[i];
  const float4* s = (const float4*)(struc + (size_t)g * INFO);
  float4* d = (float4*)(xin + (size_t)i * XCOLS + HID);  // 256B-offset, aligned
#pragma unroll
  for (int q = 0; q < INFO / 4; ++q) d[q] = s[q];
  atomicAdd(&counts[g], 1);
}

// ---------------------------------------------------------------------------
// [N,K] x [K,64] GEMM via V_WMMA_F32_16X16X4_F32 (wave32; one wave per 16x16
// output tile; 4 waves/block cover the 64 output columns of a 16-row strip).
// MODE 0: A = concat(x[N,128], xin[:,64:96])          (K=160, layer-0 input)
// MODE 1: A = relu(z*scale + shift)                   (K=64,  hidden BN fused)
// MODE 2: A = xin + agg                               (K=96,  GIN aggregate)
// Adds bias, stores Z=[N,64], atomically accumulates per-col sum / sumsq.
template <int MODE, int K>
__global__ __launch_bounds__(128)
void k_gemm64_wmma(const float* __restrict__ A0, const float* __restrict__ A1,
                   const float* __restrict__ scale, const float* __restrict__ shift,
                   const float* __restrict__ W,     // [K,64] row-major
                   const float* __restrict__ bias,  // [64]
                   float* __restrict__ Z,           // [N,64]
                   float* __restrict__ sums, float* __restrict__ sumsq) {
  __shared__ float sW[K * HID];        // <= 40 KB (K=160); WGP has 320 KB
  __shared__ float sS[HID];
  __shared__ float sT[HID];

  const int tid = threadIdx.x;
  // Cooperative stage of the weight panel into LDS (float4 copies).
  {
    const float4* W4 = (const float4*)W;
    float4* sW4 = (float4*)sW;
#pragma unroll
    for (int i = tid; i < K * HID / 4; i += 128) sW4[i] = W4[i];
  }
  if (MODE == 1) {
    if (tid < HID) sS[tid] = scale[tid];
    else           sT[tid - HID] = shift[tid - HID];
  }
  __syncthreads();

  const int lane = tid & 31;
  const int wave = tid >> 5;
  const int m0   = (int)blockIdx.x * 16;
  const int rlo  = lane & 15;
  const int khi  = (lane >> 4) * 2;   // per-lane K sub-offset (ISA A/B layout)
  const int col  = wave * 16 + rlo;
  const int row  = m0 + rlo;

  v8f c = {0.f, 0.f, 0.f, 0.f, 0.f, 0.f, 0.f, 0.f};

#pragma unroll
  for (int kb = 0; kb < K; kb += 4) {
    const int k = kb + khi;
    v2f a;
    if (MODE == 0) {
      if (kb < 128) {                 // compile-time after unroll
        const float2 p = *(const float2*)(A0 + (size_t)row * 128 + k);
        a.x = p.x; a.y = p.y;
      } else {
        const float2 p = *(const float2*)(A1 + (size_t)row * XCOLS + HID + (k - 128));
        a.x = p.x; a.y = p.y;
      }
    } else if (MODE == 1) {
      const float2 z = *(const float2*)(A0 + (size_t)row * HID + k);
      a.x = fmaxf(fmaf(z.x, sS[k],     sT[k]),     0.f);
      a.y = fmaxf(fmaf(z.y, sS[k + 1], sT[k + 1]), 0.f);
    } else {
      const float2 p = *(const float2*)(A0 + (size_t)row * XCOLS + k);
      const float2 q = *(const float2*)(A1 + (size_t)row * XCOLS + k);
      a.x = p.x + q.x; a.y = p.y + q.y;
    }
    v2f b;
    b.x = sW[k * HID + col];
    b.y = sW[(k + 1) * HID + col];
    c = __builtin_amdgcn_wmma_f32_16x16x4_f32(
        /*neg_a=*/false, a, /*neg_b=*/false, b,
        /*c_mod=*/(short)0, c, /*reuse_a=*/false, /*reuse_b=*/false);
  }

  const float bc = bias[col];
  float s = 0.f, sq = 0.f;
#pragma unroll
  for (int r = 0; r < 8; ++r) {
    float v = c[r] + bc;
    int m = m0 + r + 8 * (lane >> 4);  // C/D layout: lanes 16-31 -> M=r+8
    Z[(size_t)m * HID + col] = v;
    s += v; sq += v * v;
  }
  atomicAdd(&sums[col], s);
  atomicAdd(&sumsq[col], sq);
}

// ---------------------------------------------------------------------------
// Fold column sums into BN scale/shift: y = z*scale + shift.
__global__ void k_bn_finalize(const float* __restrict__ sums,
                              const float* __restrict__ sumsq,
                              const float* __restrict__ g,
                              const float* __restrict__ be,
                              float* __restrict__ scale,
                              float* __restrict__ shift, float invN) {
  int c = threadIdx.x;                // 64 threads
  float m  = sums[c] * invN;
  float var = sumsq[c] * invN - m * m;
  float sc = g[c] * rsqrtf(var + BN_EPS);
  scale[c] = sc;
  shift[c] = be[c] - m * sc;
}

// ---------------------------------------------------------------------------
// h = relu(BN(z2)); write into xin[:,0:64]; pool per graph.
__global__ void k_finalize_h(const float* __restrict__ z2,
                             const float* __restrict__ scale,
                             const float* __restrict__ shift,
                             const int* __restrict__ batch,
                             float* __restrict__ xin,
                             float* __restrict__ pooled, int N) {
  int idx = blockIdx.x * blockDim.x + threadIdx.x;
  if (idx >= N * HID) return;
  int i = idx >> 6, f = idx & 63;
  float h = fmaxf(fmaf(z2[idx], scale[f], shift[f]), 0.f);
  xin[(size_t)i * XCOLS + f] = h;
  atomicAdd(&pooled[batch[i] * HID + f], h);
}

// ---------------------------------------------------------------------------
// agg[dst] += xin[src]  (96 floats per edge; 24 threads/edge, float4 gathers)
__global__ void k_edge_agg(const int* __restrict__ src,
                           const int* __restrict__ dst,
                           const float* __restrict__ xin,
                           float* __restrict__ agg, int E) {
  int idx = blockIdx.x * blockDim.x + threadIdx.x;
  if (idx >= E * 24) return;
  int e = idx / 24, q = idx % 24;
  int s = src[e], d = dst[e];
  const float4 v = *(const float4*)(xin + (size_t)s * XCOLS + q * 4);
  float* a = agg + (size_t)d * XCOLS + q * 4;
  atomicAdd(a + 0, v.x); atomicAdd(a + 1, v.y);
  atomicAdd(a + 2, v.z); atomicAdd(a + 3, v.w);
}

// ---------------------------------------------------------------------------
// out[g] += concat(pooled[g], cnt*struc[g]) @ W + cnt*b   (cnt=1 for convs)
__global__ void k_linear_out(const float* __restrict__ pooled,
                             const float* __restrict__ struc,
                             const int* __restrict__ counts,
                             const float* __restrict__ W,  // [96,10]
                             const float* __restrict__ b,  // [10]
                             float* __restrict__ out, int use_counts) {
  int idx = blockIdx.x * blockDim.x + threadIdx.x;
  if (idx >= NGRAPH * TARGET) return;
  int g = idx / TARGET, t = idx % TARGET;
  float cnt = use_counts ? (float)counts[g] : 1.f;
  float acc = cnt * b[t];
  const float* pg = pooled + g * HID;
  for (int f = 0; f < HID; ++f)  acc += pg[f] * W[f * TARGET + t];
  const float* sg = struc + g * INFO;
  for (int i = 0; i < INFO; ++i) acc += cnt * sg[i] * W[(HID + i) * TARGET + t];
  out[idx] += acc;
}

// ---------------------------------------------------------------------------
extern "C" void kernel_launch(void* const* d_in, const int* in_sizes, int n_in,
                              void* d_out, int out_size, void* d_ws, size_t ws_size,
                              hipStream_t stream) {
  const float* x     = (const float*)d_in[0];
  const int*   ei    = (const int*)d_in[1];
  const int*   batch = (const int*)d_in[2];
  const float* struc = (const float*)d_in[3];
  const int E = in_sizes[1] / 2;
  const int N = in_sizes[2];           // 100000, multiple of 16

  // first_h params at 4..11; nns[l] at 12+8l; linears[l] at 44+2l
  const float* fh[8];
  for (int i = 0; i < 8; ++i) fh[i] = (const float*)d_in[4 + i];
  const float* nn[4][8];
  for (int l = 0; l < 4; ++l)
    for (int i = 0; i < 8; ++i) nn[l][i] = (const float*)d_in[12 + 8 * l + i];
  const float* linW[5]; const float* linB[5];
  for (int l = 0; l < 5; ++l) {
    linW[l] = (const float*)d_in[44 + 2 * l];
    linB[l] = (const float*)d_in[45 + 2 * l];
  }
  float* out = (float*)d_out;

  // --- workspace carve-up (256B aligned) ---
  char* ws = (char*)d_ws;
  size_t off = 0;
  auto alloc = [&](size_t bytes) {
    void* p = ws + off;
    off = (off + bytes + 255) & ~(size_t)255;
    return p;
  };
  float* xin    = (float*)alloc((size_t)N * XCOLS * 4);
  float* agg    = (float*)alloc((size_t)N * XCOLS * 4);
  float* z1     = (float*)alloc((size_t)N * HID * 4);
  float* z2     = (float*)alloc((size_t)N * HID * 4);
  float* pooled = (float*)alloc((size_t)NGRAPH * HID * 4);
  float* sum1   = (float*)alloc(128 * 4);  // sum1[64] sq1[64] contiguous
  float* sq1    = sum1 + 64;
  float* scale1 = (float*)alloc(128 * 4);
  float* shift1 = scale1 + 64;
  float* sum2   = (float*)alloc(128 * 4);
  float* sq2    = sum2 + 64;
  float* scale2 = (float*)alloc(128 * 4);
  float* shift2 = scale2 + 64;
  int*   counts = (int*)alloc(NGRAPH * 4);
  (void)ws_size;

  const float invN = 1.f / (float)N;
  const int gemmGrid = N / 16;

  // second GEMM of every MLP: A = relu(BN(z1)), K = 64
  auto run_mlp2 = [&](const float* const* p) {
    k_bn_finalize<<<1, 64, 0, stream>>>(sum1, sq1, p[2], p[3], scale1, shift1, invN);
    hipMemsetAsync(sum2, 0, 128 * 4, stream);
    k_gemm64_wmma<1, HID><<<gemmGrid, 128, 0, stream>>>(
        z1, nullptr, scale1, shift1, p[4], p[5], z2, sum2, sq2);
    k_bn_finalize<<<1, 64, 0, stream>>>(sum2, sq2, p[6], p[7], scale2, shift2, invN);
  };

  // ---- layer 0 ----
  hipMemsetAsync(counts, 0, NGRAPH * 4, stream);
  k_init_rep_counts<<<(N + 255) / 256, 256, 0, stream>>>(batch, struc, xin, counts, N);
  hipMemsetAsync(sum1, 0, 128 * 4, stream);
  k_gemm64_wmma<0, 160><<<gemmGrid, 128, 0, stream>>>(
      x, xin, nullptr, nullptr, fh[0], fh[1], z1, sum1, sq1);
  run_mlp2(fh);
  hipMemsetAsync(pooled, 0, (size_t)NGRAPH * HID * 4, stream);
  k_finalize_h<<<(N * HID + 255) / 256, 256, 0, stream>>>(z2, scale2, shift2,
                                                          batch, xin, pooled, N);
  hipMemsetAsync(out, 0, (size_t)NGRAPH * TARGET * 4, stream);
  k_linear_out<<<(NGRAPH * TARGET + 255) / 256, 256, 0, stream>>>(
      pooled, struc, counts, linW[0], linB[0], out, /*use_counts=*/1);

  // ---- conv layers ----
  for (int l = 0; l < 4; ++l) {
    hipMemsetAsync(agg, 0, (size_t)N * XCOLS * 4, stream);
    int tE = E * 24;
    k_edge_agg<<<(tE + 255) / 256, 256, 0, stream>>>(ei, ei + E, xin, agg, E);
    hipMemsetAsync(sum1, 0, 128 * 4, stream);
    k_gemm64_wmma<2, XCOLS><<<gemmGrid, 128, 0, stream>>>(
        xin, agg, nullptr, nullptr, nn[l][0], nn[l][1], z1, sum1, sq1);
    run_mlp2(nn[l]);
    hipMemsetAsync(pooled, 0, (size_t)NGRAPH * HID * 4, stream);
    k_finalize_h<<<(N * HID + 255) / 256, 256, 0, stream>>>(z2, scale2, shift2,
                                                            batch, xin, pooled, N);
    k_linear_out<<<(NGRAPH * TARGET + 255) / 256, 256, 0, stream>>>(
        pooled, struc, counts, linW[l + 1], linB[l + 1], out, /*use_counts=*/0);
  }
}